// AttnDecoderRNN_20263655702618
// MI455X (gfx1250) — compile-verified
//
#include <hip/hip_runtime.h>
#include <math.h>

// ---------------------------------------------------------------------------
// AttnDecoderRNN single step on gfx1250 (CDNA5, wave32).
// All heavy work is fp32 GEMV -> HBM-bandwidth bound (~240 MB of weights,
// ~10 us floor at 23.3 TB/s).  GEMVs run on the matrix pipe via
// V_WMMA_F32_16X16X4_F32 (exact fp32), one 16-row tile per wave, x staged in
// LDS.  K is permuted per-wmma so each lane issues one contiguous b128 load
// per 8 K-values; 8 loads are batched per iteration for deep per-wave MLP.
// ---------------------------------------------------------------------------

constexpr int HDIM  = 1024;
constexpr int VOCAB = 50257;
constexpr int LEN   = 10;

typedef __attribute__((ext_vector_type(2))) float v2f;
typedef __attribute__((ext_vector_type(4))) float v4f;
typedef __attribute__((ext_vector_type(8))) float v8f;

// y[r] = dot(W[r,:], x) + bias[r]  (optional relu).  W row-major [R,K],
// K multiple of 64, K <= 2048.  One 16-row tile per wave32.
__global__ __launch_bounds__(256) void wmma_gemv_f32(
    const float* __restrict__ W, const float* __restrict__ x,
    const float* __restrict__ bias, float* __restrict__ y,
    int R, int K, int relu)
{
    __shared__ float xs[2048];
    const int tid = threadIdx.x;
    for (int i = tid; i < K; i += 256) xs[i] = x[i];   // x may be only 4B-aligned
    __syncthreads();

    const int wave    = tid >> 5;
    const int lane    = tid & 31;
    const int tile    = blockIdx.x * 8 + wave;
    const int rowbase = tile << 4;
    if (rowbase >= R) return;                   // wave-uniform: EXEC stays all-ones

    const int mrow  = rowbase + (lane & 15);
    const int klane = (lane >> 4) << 2;         // lanes 0-15: +0, lanes 16-31: +4
    const int r     = (mrow < R) ? mrow : (R - 1);   // clamp tail rows (safe reads)
    const float* __restrict__ Wrow = W + (size_t)r * (size_t)K + klane;
    const float* xrow = xs + klane;

    v8f acc0 = {0.f,0.f,0.f,0.f,0.f,0.f,0.f,0.f};
    v8f acc1 = {0.f,0.f,0.f,0.f,0.f,0.f,0.f,0.f};

    // 64 K-values per iteration: batch 8 global b128 + 8 LDS b128 into distinct
    // registers first (deep MLP), then run the 16-wmma block.
    for (int k = 0; k < K; k += 64) {
        v4f a[8], b[8];
        #pragma unroll
        for (int u = 0; u < 8; ++u)
            a[u] = *(const v4f*)(Wrow + k + 8 * u);  // 16B aligned
        #pragma unroll
        for (int u = 0; u < 8; ++u)
            b[u] = *(const v4f*)(xrow + k + 8 * u);
        #pragma unroll
        for (int u = 0; u < 8; ++u) {
            // lane<16 holds K {k..k+3}, lane>=16 {k+4..k+7}; wmma0 takes .xy
            // (K-set {k,k+1,k+4,k+5}), wmma1 takes .zw — K-order is irrelevant
            // for a dot product as long as A and B use the same permutation.
            v2f a0 = __builtin_shufflevector(a[u], a[u], 0, 1);
            v2f a1 = __builtin_shufflevector(a[u], a[u], 2, 3);
            v2f b0 = __builtin_shufflevector(b[u], b[u], 0, 1);
            v2f b1 = __builtin_shufflevector(b[u], b[u], 2, 3);
            // every B column == x chunk, so every D column == W·x
            acc0 = __builtin_amdgcn_wmma_f32_16x16x4_f32(false, a0, false, b0,
                                                         (short)0, acc0, false, false);
            acc1 = __builtin_amdgcn_wmma_f32_16x16x4_f32(false, a1, false, b1,
                                                         (short)0, acc1, false, false);
        }
    }

    // D layout: lane 0 (N=0) has rows base+0..7 in v[0..7]; lane 16 rows base+8..15
    if ((lane & 15) == 0) {
        const int mbase = rowbase + ((lane >> 4) << 3);
        #pragma unroll
        for (int i = 0; i < 8; ++i) {
            const int rr = mbase + i;
            if (rr < R) {
                float v = acc0[i] + acc1[i] + bias[rr];
                if (relu) v = fmaxf(v, 0.0f);
                y[rr] = v;
            }
        }
    }
}

// Embedding lookup + attention softmax + context; writes xin=[embedded|context]
// and the 10 attention weights.  Single block.
__global__ __launch_bounds__(256) void attn_kernel(
    const int* __restrict__ input, const float* __restrict__ hidden,
    const float* __restrict__ enc, const float* __restrict__ emb,
    const float* __restrict__ attn_W, const float* __restrict__ attn_b,
    float* __restrict__ xin, float* __restrict__ attnw_out)
{
    __shared__ float red[256];
    __shared__ float w[LEN];
    const int tid   = threadIdx.x;
    const int token = input[0];

    for (int i = tid; i < HDIM; i += 256)
        xin[i] = emb[(size_t)token * HDIM + i];
    __syncthreads();

    // attn logits over attn_in = [embedded | hidden]
    for (int j = 0; j < LEN; ++j) {
        const float* wr = attn_W + j * (2 * HDIM);
        float p = 0.f;
        for (int t = tid; t < 2 * HDIM; t += 256) {
            float v = (t < HDIM) ? xin[t] : hidden[t - HDIM];
            p = fmaf(v, wr[t], p);
        }
        red[tid] = p; __syncthreads();
        for (int s = 128; s > 0; s >>= 1) {
            if (tid < s) red[tid] += red[tid + s];
            __syncthreads();
        }
        if (tid == 0) w[j] = red[0] + attn_b[j];
        __syncthreads();
    }

    // softmax over 10 (serial on thread 0, trivial)
    if (tid == 0) {
        float m = w[0];
        for (int j = 1; j < LEN; ++j) m = fmaxf(m, w[j]);
        float s = 0.f;
        for (int j = 0; j < LEN; ++j) { w[j] = expf(w[j] - m); s += w[j]; }
        const float inv = 1.f / s;
        for (int j = 0; j < LEN; ++j) w[j] *= inv;
    }
    __syncthreads();
    if (tid < LEN) attnw_out[tid] = w[tid];

    // context = attn_weights @ encoder_outputs
    for (int i = tid; i < HDIM; i += 256) {
        float s = 0.f;
        #pragma unroll
        for (int j = 0; j < LEN; ++j) s = fmaf(w[j], enc[j * HDIM + i], s);
        xin[HDIM + i] = s;
    }
}

// GRU gates (PyTorch ordering r,z,n) -> h_new
__global__ __launch_bounds__(256) void gru_gates_kernel(
    const float* __restrict__ gi, const float* __restrict__ gh,
    const float* __restrict__ h, float* __restrict__ hnew)
{
    const int i = blockIdx.x * 256 + threadIdx.x;
    if (i < HDIM) {
        const float r = 1.f / (1.f + expf(-(gi[i] + gh[i])));
        const float z = 1.f / (1.f + expf(-(gi[HDIM + i] + gh[HDIM + i])));
        const float n = tanhf(gi[2 * HDIM + i] + r * gh[2 * HDIM + i]);
        hnew[i] = (1.f - z) * n + z * h[i];
    }
}

// In-place log-softmax over n logits, single block (L2-resident, cheap).
__global__ __launch_bounds__(1024) void logsoftmax_kernel(
    float* __restrict__ logits, int n)
{
    __shared__ float red[1024];
    const int tid = threadIdx.x;

    float m = -3.402823466e38f;
    for (int i = tid; i < n; i += 1024) m = fmaxf(m, logits[i]);
    red[tid] = m; __syncthreads();
    for (int s = 512; s > 0; s >>= 1) {
        if (tid < s) red[tid] = fmaxf(red[tid], red[tid + s]);
        __syncthreads();
    }
    const float gm = red[0]; __syncthreads();

    float p = 0.f;
    for (int i = tid; i < n; i += 1024) p += expf(logits[i] - gm);
    red[tid] = p; __syncthreads();
    for (int s = 512; s > 0; s >>= 1) {
        if (tid < s) red[tid] += red[tid + s];
        __syncthreads();
    }
    const float lse = gm + logf(red[0]);
    __syncthreads();

    for (int i = tid; i < n; i += 1024) logits[i] -= lse;
}

extern "C" void kernel_launch(void* const* d_in, const int* in_sizes, int n_in,
                              void* d_out, int out_size, void* d_ws, size_t ws_size,
                              hipStream_t stream)
{
    const int*   input   = (const int*)  d_in[0];
    const float* hidden  = (const float*)d_in[1];   // [1,1,H]
    const float* enc     = (const float*)d_in[2];   // [L,H]
    const float* emb     = (const float*)d_in[3];   // [V,H]
    const float* attn_W  = (const float*)d_in[4];   // [L,2H]
    const float* attn_b  = (const float*)d_in[5];   // [L]
    const float* comb_W  = (const float*)d_in[6];   // [H,2H]
    const float* comb_b  = (const float*)d_in[7];   // [H]
    const float* gru_wih = (const float*)d_in[8];   // [3H,H]
    const float* gru_whh = (const float*)d_in[9];   // [3H,H]
    const float* gru_bih = (const float*)d_in[10];  // [3H]
    const float* gru_bhh = (const float*)d_in[11];  // [3H]
    const float* out_W   = (const float*)d_in[12];  // [V,H]
    const float* out_b   = (const float*)d_in[13];  // [V]

    float* out   = (float*)d_out;
    float* logp  = out;                  // [V]   log_probs (logits in-place)
    float* hnew  = out + VOCAB;          // [H]   h_new
    float* attnw = out + VOCAB + HDIM;   // [L]   attn_weights

    float* ws   = (float*)d_ws;
    float* xin  = ws;                    // 2048: [embedded | context]
    float* xcmb = ws + 2048;             // 1024: relu(comb)
    float* gi   = ws + 3072;             // 3072
    float* gh   = ws + 6144;             // 3072

    // 1) attention (tiny, one workgroup)
    attn_kernel<<<1, 256, 0, stream>>>(input, hidden, enc, emb,
                                       attn_W, attn_b, xin, attnw);
    // 2) x = relu(comb_W @ xin + comb_b)        [1024 x 2048]
    wmma_gemv_f32<<<8, 256, 0, stream>>>(comb_W, xin, comb_b, xcmb,
                                         HDIM, 2 * HDIM, 1);
    // 3) gi = gru_wih @ x + bih                 [3072 x 1024]
    wmma_gemv_f32<<<24, 256, 0, stream>>>(gru_wih, xcmb, gru_bih, gi,
                                          3 * HDIM, HDIM, 0);
    // 4) gh = gru_whh @ h + bhh                 [3072 x 1024]
    wmma_gemv_f32<<<24, 256, 0, stream>>>(gru_whh, hidden, gru_bhh, gh,
                                          3 * HDIM, HDIM, 0);
    // 5) gates -> h_new
    gru_gates_kernel<<<(HDIM + 255) / 256, 256, 0, stream>>>(gi, gh, hidden, hnew);
    // 6) logits = out_W @ h_new + out_b         [50257 x 1024]  (dominant, ~206 MB)
    {
        const int tiles  = (VOCAB + 15) / 16;
        const int blocks = (tiles + 7) / 8;
        wmma_gemv_f32<<<blocks, 256, 0, stream>>>(out_W, hnew, out_b, logp,
                                                  VOCAB, HDIM, 0);
    }
    // 7) in-place log-softmax
    logsoftmax_kernel<<<1, 1024, 0, stream>>>(logp, VOCAB);
}